// Model_5755256177322
// MI455X (gfx1250) — compile-verified
//
#include <hip/hip_runtime.h>
#include <hip/hip_bf16.h>

#define BB 16384
#define NT 50
#define DT_C 0.5f
#define SQDT_C 0.7071067811865476f

typedef __attribute__((ext_vector_type(2))) float v2f;
typedef __attribute__((ext_vector_type(8))) float v8f;

__device__ __forceinline__ void wave_sync_lds() {
  // intra-wave LDS ordering: ds ops are in-order in HW; this stops compiler reordering
  asm volatile("s_wait_dscnt 0x0" ::: "memory");
}

__device__ __forceinline__ float elu1(float x)      { return x > 0.f ? x : expm1f(x); }
__device__ __forceinline__ float softplus1(float x) { return x > 20.f ? x : log1pf(expf(x)); }
__device__ __forceinline__ float sigmoid1(float x)  { return 1.f / (1.f + expf(-x)); }

// C(16x16) += A(16xK) @ W(NxK)^T  (subtile columns n0..n0+15), K multiple of 4.
// fp32 WMMA: A frag = v2f per lane (lanes 0-15: K=k0,k0+1 ; lanes 16-31: K=k0+2,k0+3),
// B frag mirrored with N across lanes.
__device__ __forceinline__ v8f wmma_tile(const float* A, int lda,
                                         const float* W, int ldw,
                                         int n0, int K, v8f acc) {
  const int lane = threadIdx.x & 31;
  const int hlf  = lane >> 4;
  const int idx  = lane & 15;
  const float* arow = A + idx * lda;
  const float* wrow = W + (n0 + idx) * ldw;
  for (int k0 = 0; k0 < K; k0 += 4) {
    v2f a, b;
    a.x = arow[k0 + 2*hlf];
    a.y = arow[k0 + 2*hlf + 1];
    b.x = wrow[k0 + 2*hlf];
    b.y = wrow[k0 + 2*hlf + 1];
    acc = __builtin_amdgcn_wmma_f32_16x16x4_f32(false, a, false, b, (short)0, acc,
                                                false, false);
  }
  return acc;
}

// ---------------- Encoder stage 1: enc_in -> h1_pre (B x 95) ----------------
__global__ __launch_bounds__(256) void enc1_kernel(
    const float* __restrict__ Y, const float* __restrict__ times,
    const float* __restrict__ mask0, const float* __restrict__ env,
    const float* __restrict__ med, const float* __restrict__ missing,
    const float* __restrict__ pop_std, const float* __restrict__ eps_y0,
    const float* __restrict__ We1, const float* __restrict__ be1,
    float* __restrict__ h1pre)
{
  __shared__ float sW[95*95];
  __shared__ float sb[95];
  for (int i = threadIdx.x; i < 95*95; i += 256) sW[i] = We1[i];
  for (int i = threadIdx.x; i < 95;    i += 256) sb[i] = be1[i];
  __syncthreads();
  const int b = blockIdx.x*256 + threadIdx.x;
  float xin[95];
  for (int n = 0; n < 29; ++n) {
    float m = mask0[b*29+n];
    float y = Y[b*580 + n];  // Y[:,0,:]
    xin[n]    = m*y + (1.f-m)*(missing[b*29+n] + pop_std[b*29+n]*eps_y0[b*29+n]);
    xin[29+n] = m;
  }
  xin[58] = (times[b*20] - 70.f) / 7.f;
  for (int n = 0; n < 26; ++n) xin[59+n] = env[b*26+n];
  for (int n = 0; n < 10; ++n) xin[85+n] = med[b*200+n];  // med[:,0,:]
  for (int j = 0; j < 95; ++j) {
    float a = sb[j];
    const float* w = &sW[j*95];
    for (int k = 0; k < 95; ++k) a += w[k]*xin[k];
    h1pre[b*95+j] = a;
  }
}

// ---------------- BN stats: one block per feature -> scale/shift ----------------
__global__ __launch_bounds__(256) void bnstats_kernel(
    const float* __restrict__ X, int F,
    const float* __restrict__ g, const float* __restrict__ bt,
    float* __restrict__ scale, float* __restrict__ shift)
{
  __shared__ float ssum[256];
  __shared__ float ssq[256];
  const int j = blockIdx.x;
  float s = 0.f, q = 0.f;
  for (int i = threadIdx.x; i < BB; i += 256) {
    float v = X[i*F + j];
    s += v; q += v*v;
  }
  ssum[threadIdx.x] = s; ssq[threadIdx.x] = q;
  __syncthreads();
  for (int off = 128; off > 0; off >>= 1) {
    if (threadIdx.x < off) {
      ssum[threadIdx.x] += ssum[threadIdx.x+off];
      ssq [threadIdx.x] += ssq [threadIdx.x+off];
    }
    __syncthreads();
  }
  if (threadIdx.x == 0) {
    float mean = ssum[0] / (float)BB;
    float var  = ssq[0] / (float)BB - mean*mean;
    float sc   = g[j] * rsqrtf(var + 1e-5f);
    scale[j] = sc;
    shift[j] = bt[j] - mean*sc;
  }
}

// ---------------- Encoder stage 2: bn1+elu, -> h2_pre (B x 70) ----------------
__global__ __launch_bounds__(256) void enc2_kernel(
    const float* __restrict__ h1pre, const float* __restrict__ sc1,
    const float* __restrict__ sh1, const float* __restrict__ We2,
    const float* __restrict__ be2, float* __restrict__ h2pre)
{
  __shared__ float sW[70*95];
  __shared__ float sb[70];
  __shared__ float ssc[95];
  __shared__ float ssh[95];
  for (int i = threadIdx.x; i < 70*95; i += 256) sW[i] = We2[i];
  for (int i = threadIdx.x; i < 70;    i += 256) sb[i] = be2[i];
  for (int i = threadIdx.x; i < 95;    i += 256) { ssc[i] = sc1[i]; ssh[i] = sh1[i]; }
  __syncthreads();
  const int b = blockIdx.x*256 + threadIdx.x;
  float h[95];
  for (int k = 0; k < 95; ++k) h[k] = elu1(h1pre[b*95+k]*ssc[k] + ssh[k]);
  for (int j = 0; j < 70; ++j) {
    float a = sb[j];
    const float* w = &sW[j*95];
    for (int k = 0; k < 95; ++k) a += w[k]*h[k];
    h2pre[b*70+j] = a;
  }
}

// -------- Encoder stage 3: bn2+elu -> mu0/logvar0/z0, ctx, LSTM init (hh0,cc0) --------
__global__ __launch_bounds__(256) void enc3_kernel(
    const float* __restrict__ h2pre, const float* __restrict__ sc2,
    const float* __restrict__ sh2, const float* __restrict__ We3,
    const float* __restrict__ be3, const float* __restrict__ eps_z,
    const float* __restrict__ env, const float* __restrict__ times,
    const float* __restrict__ Wm, const float* __restrict__ bm,
    float* __restrict__ mu0o, float* __restrict__ lv0o, float* __restrict__ ctxo,
    float* __restrict__ z0w, float* __restrict__ hh0w, float* __restrict__ cc0w)
{
  __shared__ float sW3[40*70];
  __shared__ float sb3[40];
  __shared__ float sWm[50*24];
  __shared__ float sbm[50];
  __shared__ float ssc[70];
  __shared__ float ssh[70];
  for (int i = threadIdx.x; i < 40*70; i += 256) sW3[i] = We3[i];
  for (int i = threadIdx.x; i < 40;    i += 256) sb3[i] = be3[i];
  for (int i = threadIdx.x; i < 50*24; i += 256) sWm[i] = Wm[i];
  for (int i = threadIdx.x; i < 50;    i += 256) sbm[i] = bm[i];
  for (int i = threadIdx.x; i < 70;    i += 256) { ssc[i] = sc2[i]; ssh[i] = sh2[i]; }
  __syncthreads();
  const int b = blockIdx.x*256 + threadIdx.x;
  float h[70];
  for (int k = 0; k < 70; ++k) h[k] = elu1(h2pre[b*70+k]*ssc[k] + ssh[k]);
  float mu[20];
  float min_[24];  // [tt0, z0(20), ctx(3)]
  min_[0] = (times[b*20] - 70.f) / 7.f;
  for (int j = 0; j < 40; ++j) {
    float a = sb3[j];
    const float* w = &sW3[j*70];
    for (int k = 0; k < 70; ++k) a += w[k]*h[k];
    if (j < 20) { mu[j] = a; mu0o[b*20+j] = a; }
    else {
      int jj = j - 20;
      lv0o[b*20+jj] = a;
      float z = mu[jj] + expf(0.5f*a)*eps_z[b*20+jj];
      z0w[b*20+jj] = z;
      min_[1+jj] = z;
    }
  }
  float c0 = env[b*26+8], c1 = env[b*26+12], c2 = env[b*26+13];
  ctxo[b*3] = c0; ctxo[b*3+1] = c1; ctxo[b*3+2] = c2;
  min_[21] = c0; min_[22] = c1; min_[23] = c2;
  for (int j = 0; j < 50; ++j) {
    float a = sbm[j];
    const float* w = &sWm[j*24];
    for (int k = 0; k < 24; ++k) a += w[k]*min_[k];
    a = tanhf(a);
    if (j < 25) hh0w[b*25+j] = a;
    else        cc0w[b*25+(j-25)] = a;
  }
}

// ---------------- SDE scan: 4 waves/block, 16 batch rows per wave, WMMA f32 ----------------
__global__ __launch_bounds__(128) void sde_kernel(
    const float* __restrict__ z0, const float* __restrict__ hh0,
    const float* __restrict__ cc0, const float* __restrict__ times,
    const float* __restrict__ ctxg, const float* __restrict__ eps,
    const float* __restrict__ Wf1, const float* __restrict__ bf1,
    const float* __restrict__ Wf2, const float* __restrict__ bf2,
    const float* __restrict__ Wsp, const float* __restrict__ bsp,
    const float* __restrict__ Wi,  const float* __restrict__ Wh,
    const float* __restrict__ bi,  const float* __restrict__ bh,
    const float* __restrict__ Wg,  const float* __restrict__ bg,
    float* __restrict__ predZ, float* __restrict__ tall,
    float* __restrict__ predS, float* __restrict__ predLG,
    float* __restrict__ predSig)
{
  __shared__ float sWf1[64*24];    // 64 x 24
  __shared__ float sWf2[32*64];    // pad N 20->32
  __shared__ float sWs [32*24];    // pad N 20->32
  __shared__ float sWi [112*24];   // pad N 100->112
  __shared__ float sWh [112*28];   // pad N 100->112, K 25->28
  __shared__ float sWg [25];
  __shared__ float sbf1[64];
  __shared__ float sbf2[32];
  __shared__ float sbs [32];
  __shared__ float sbih[112];
  __shared__ float sbg [1];
  __shared__ float sInp[4][16*24];
  __shared__ float sHH [4][16*28];
  __shared__ float sCC [4][16*25];
  __shared__ float sGt [4][16*112];
  __shared__ float sF1 [4][16*64];
  __shared__ float sDX [4][16*32];
  __shared__ float sSG [4][16*32];
  __shared__ float sS  [4][16];

  const int tid = threadIdx.x;
  for (int i = tid; i < 64*24;  i += 128) sWf1[i] = Wf1[i];
  for (int i = tid; i < 32*64;  i += 128) { int n = i>>6;  sWf2[i] = (n<20)  ? Wf2[i] : 0.f; }
  for (int i = tid; i < 32*24;  i += 128) { int n = i/24;  sWs[i]  = (n<20)  ? Wsp[i] : 0.f; }
  for (int i = tid; i < 112*24; i += 128) { int n = i/24;  sWi[i]  = (n<100) ? Wi[i]  : 0.f; }
  for (int i = tid; i < 112*28; i += 128) { int n = i/28, k = i%28;
                                            sWh[i] = (n<100 && k<25) ? Wh[n*25+k] : 0.f; }
  for (int i = tid; i < 25;  i += 128) sWg[i]  = Wg[i];
  for (int i = tid; i < 64;  i += 128) sbf1[i] = bf1[i];
  for (int i = tid; i < 32;  i += 128) sbf2[i] = (i<20)  ? bf2[i] : 0.f;
  for (int i = tid; i < 32;  i += 128) sbs[i]  = (i<20)  ? bsp[i] : 0.f;
  for (int i = tid; i < 112; i += 128) sbih[i] = (i<100) ? (bi[i]+bh[i]) : 0.f;
  if (tid == 0) sbg[0] = bg[0];
  __syncthreads();

  const int wave = tid >> 5;
  const int lane = tid & 31;
  const int hlf  = lane >> 4;
  const int idx  = lane & 15;
  const int rowbase = blockIdx.x*64 + wave*16;
  float* inp = sInp[wave];
  float* hh  = sHH[wave];
  float* cc  = sCC[wave];
  float* gt  = sGt[wave];
  float* f1  = sF1[wave];
  float* dx  = sDX[wave];
  float* sg  = sSG[wave];
  float* Sv  = sS[wave];

  for (int e = lane; e < 16*24; e += 32) {
    int m = e/24, k = e%24;
    float v = 0.f;
    if (k < 20)       v = z0[(rowbase+m)*20 + k];
    else if (k >= 21) v = ctxg[(rowbase+m)*3 + (k-21)];
    inp[e] = v;
  }
  for (int e = lane; e < 16*28; e += 32) {
    int m = e/28, k = e%28;
    hh[e] = (k < 25) ? hh0[(rowbase+m)*25 + k] : 0.f;
  }
  for (int e = lane; e < 16*25; e += 32)
    cc[e] = cc0[(rowbase + e/25)*25 + (e%25)];
  if (lane < 16) Sv[lane] = 1.f;
  const float t0r = (lane < 16) ? times[(rowbase+lane)*20] : 0.f;
  wave_sync_lds();

  for (int s = 0; s < NT; ++s) {
    if (lane < 16) {
      float t = t0r + DT_C * (float)s;
      inp[lane*24 + 20] = (t - 70.f) / 7.f;
      tall[(rowbase+lane)*NT + s] = t;
    }
    for (int e = lane; e < 16*20; e += 32) {
      int m = e/20, k = e%20;
      predZ[((rowbase+m)*NT + s)*20 + k] = inp[m*24 + k];
    }
    wave_sync_lds();

    // sig = softplus(inp @ Ws^T + bs)
    for (int nc = 0; nc < 2; ++nc) {
      v8f acc = {0.f,0.f,0.f,0.f,0.f,0.f,0.f,0.f};
      acc = wmma_tile(inp, 24, sWs, 24, nc*16, 24, acc);
      int n = nc*16 + idx;
      float bv = sbs[n];
      #pragma unroll
      for (int r = 0; r < 8; ++r) sg[(r + 8*hlf)*32 + n] = softplus1(acc[r] + bv);
    }
    // f1h = elu(inp @ Wf1^T + bf1)
    for (int nc = 0; nc < 4; ++nc) {
      v8f acc = {0.f,0.f,0.f,0.f,0.f,0.f,0.f,0.f};
      acc = wmma_tile(inp, 24, sWf1, 24, nc*16, 24, acc);
      int n = nc*16 + idx;
      float bv = sbf1[n];
      #pragma unroll
      for (int r = 0; r < 8; ++r) f1[(r + 8*hlf)*64 + n] = elu1(acc[r] + bv);
    }
    // gates raw = inp @ Wi^T + hh @ Wh^T
    for (int nc = 0; nc < 7; ++nc) {
      v8f acc = {0.f,0.f,0.f,0.f,0.f,0.f,0.f,0.f};
      acc = wmma_tile(inp, 24, sWi, 24, nc*16, 24, acc);
      acc = wmma_tile(hh,  28, sWh, 28, nc*16, 28, acc);
      int n = nc*16 + idx;
      #pragma unroll
      for (int r = 0; r < 8; ++r) gt[(r + 8*hlf)*112 + n] = acc[r];
    }
    wave_sync_lds();
    // dx = f1h @ Wf2^T + bf2
    for (int nc = 0; nc < 2; ++nc) {
      v8f acc = {0.f,0.f,0.f,0.f,0.f,0.f,0.f,0.f};
      acc = wmma_tile(f1, 64, sWf2, 64, nc*16, 64, acc);
      int n = nc*16 + idx;
      float bv = sbf2[n];
      #pragma unroll
      for (int r = 0; r < 8; ++r) dx[(r + 8*hlf)*32 + n] = acc[r] + bv;
    }
    wave_sync_lds();

    // LSTM elementwise update
    for (int e = lane; e < 16*25; e += 32) {
      int m = e/25, j = e%25;
      const float* g = &gt[m*112];
      float iv = sigmoid1(g[j]      + sbih[j]);
      float fv = sigmoid1(g[25+j]   + sbih[25+j]);
      float gv = tanhf   (g[50+j]   + sbih[50+j]);
      float ov = sigmoid1(g[75+j]   + sbih[75+j]);
      float c2 = fv * cc[m*25+j] + iv * gv;
      cc[m*25+j] = c2;
      hh[m*28+j] = ov * tanhf(c2);
    }
    wave_sync_lds();
    // logGamma + survival
    if (lane < 16) {
      int m = lane;
      float a = sbg[0];
      for (int j = 0; j < 25; ++j) a += elu1(hh[m*28+j]) * sWg[j];
      float S2 = Sv[m] * expf(-expf(a) * DT_C);
      Sv[m] = S2;
      predLG[(rowbase+m)*NT + s] = a;
      predS [(rowbase+m)*NT + s] = S2;
    }
    // x update + sigma output
    for (int e = lane; e < 16*20; e += 32) {
      int m = e/20, k = e%20;
      int b = rowbase + m;
      float sgv = sg[m*32 + k];
      float xv = inp[m*24+k] + DT_C * dx[m*32+k]
               + sgv * SQDT_C * eps[(s*BB + b)*20 + k];
      inp[m*24+k] = xv;
      predSig[(b*NT + s)*20 + k] = sgv;
    }
    wave_sync_lds();
  }
}

// ---------------- Decoder: rows = B*NT, 24 -> 65 (LN+ELU) -> 29, WMMA f32 ----------------
__global__ __launch_bounds__(128) void dec_kernel(
    const float* __restrict__ predZ, const float* __restrict__ tall,
    const float* __restrict__ ctxg,
    const float* __restrict__ Wd1, const float* __restrict__ bd1,
    const float* __restrict__ lng, const float* __restrict__ lnb,
    const float* __restrict__ Wd2, const float* __restrict__ bd2,
    float* __restrict__ predX)
{
  __shared__ float sW1[80*24];   // pad N 65->80
  __shared__ float sb1[65];
  __shared__ float sg[65];
  __shared__ float sbb[65];
  __shared__ float sW2[32*68];   // pad N 29->32, K 65->68
  __shared__ float sb2[32];
  __shared__ float sA[4][16*24];
  __shared__ float sH[4][16*68];
  const int tid = threadIdx.x;
  for (int i = tid; i < 80*24; i += 128) { int n = i/24; sW1[i] = (n<65) ? Wd1[i] : 0.f; }
  for (int i = tid; i < 65;    i += 128) { sb1[i] = bd1[i]; sg[i] = lng[i]; sbb[i] = lnb[i]; }
  for (int i = tid; i < 32*68; i += 128) { int n = i/68, k = i%68;
                                           sW2[i] = (n<29 && k<65) ? Wd2[n*65+k] : 0.f; }
  for (int i = tid; i < 32;    i += 128) sb2[i] = (i<29) ? bd2[i] : 0.f;
  __syncthreads();

  const int wave = tid >> 5;
  const int lane = tid & 31;
  const int hlf  = lane >> 4;
  const int idx  = lane & 15;
  const int rowbase = blockIdx.x*64 + wave*16;   // rows over flattened (b, t)
  float* A = sA[wave];
  float* H = sH[wave];

  for (int e = lane; e < 16*24; e += 32) {
    int m = e/24, k = e%24;
    int r = rowbase + m;
    float v;
    if (k < 20)       v = predZ[r*20 + k];
    else if (k == 20) v = (tall[r] - 70.f) / 7.f;
    else              v = ctxg[(r/NT)*3 + (k-21)];
    A[e] = v;
  }
  wave_sync_lds();

  // hd_pre = A @ Wd1^T + bd1
  for (int nc = 0; nc < 5; ++nc) {
    v8f acc = {0.f,0.f,0.f,0.f,0.f,0.f,0.f,0.f};
    acc = wmma_tile(A, 24, sW1, 24, nc*16, 24, acc);
    int n = nc*16 + idx;
    float bv = (n < 65) ? sb1[n] : 0.f;
    #pragma unroll
    for (int r = 0; r < 8; ++r) {
      int m = r + 8*hlf;
      if (n < 65)      H[m*68 + n] = acc[r] + bv;
      else if (n < 68) H[m*68 + n] = 0.f;
    }
  }
  wave_sync_lds();
  // per-row LayerNorm(65) + ELU
  if (lane < 16) {
    float* h = &H[lane*68];
    float mean = 0.f;
    for (int k = 0; k < 65; ++k) mean += h[k];
    mean /= 65.f;
    float var = 0.f;
    for (int k = 0; k < 65; ++k) { float d = h[k]-mean; var += d*d; }
    var /= 65.f;
    float rs = rsqrtf(var + 1e-5f);
    for (int k = 0; k < 65; ++k) h[k] = elu1(sg[k]*(h[k]-mean)*rs + sbb[k]);
  }
  wave_sync_lds();
  // pred_X = hd @ Wd2^T + bd2
  for (int nc = 0; nc < 2; ++nc) {
    v8f acc = {0.f,0.f,0.f,0.f,0.f,0.f,0.f,0.f};
    acc = wmma_tile(H, 68, sW2, 68, nc*16, 68, acc);
    int n = nc*16 + idx;
    if (n < 29) {
      float bv = sb2[n];
      #pragma unroll
      for (int r = 0; r < 8; ++r)
        predX[(rowbase + r + 8*hlf)*29 + n] = acc[r] + bv;
    }
  }
}

__global__ __launch_bounds__(256) void recon_kernel(const float* __restrict__ predX,
                                                    float* __restrict__ recon)
{
  int i = blockIdx.x*256 + threadIdx.x;
  if (i < BB*29) {
    int b = i/29, n = i%29;
    recon[i] = predX[b*(NT*29) + n];   // pred_X[:,0,:]
  }
}

extern "C" void kernel_launch(void* const* d_in, const int* in_sizes, int n_in,
                              void* d_out, int out_size, void* d_ws, size_t ws_size,
                              hipStream_t stream) {
  (void)in_sizes; (void)n_in; (void)out_size; (void)ws_size;
  const float* Y       = (const float*)d_in[0];
  const float* times   = (const float*)d_in[1];
  const float* mask    = (const float*)d_in[2];
  const float* mask0   = (const float*)d_in[3];
  const float* env     = (const float*)d_in[4];
  const float* med     = (const float*)d_in[5];
  const float* missing = (const float*)d_in[6];
  const float* pop_std = (const float*)d_in[7];
  const float* surv    = (const float*)d_in[8];
  const float* dead    = (const float*)d_in[9];
  const float* after   = (const float*)d_in[10];
  const float* cens    = (const float*)d_in[11];
  const float* wts     = (const float*)d_in[12];
  const float* eps_y0  = (const float*)d_in[14];
  const float* eps_z   = (const float*)d_in[15];
  const float* eps_sde = (const float*)d_in[16];
  const float* We1 = (const float*)d_in[17]; const float* be1 = (const float*)d_in[18];
  const float* bn1g= (const float*)d_in[19]; const float* bn1b= (const float*)d_in[20];
  const float* We2 = (const float*)d_in[21]; const float* be2 = (const float*)d_in[22];
  const float* bn2g= (const float*)d_in[23]; const float* bn2b= (const float*)d_in[24];
  const float* We3 = (const float*)d_in[25]; const float* be3 = (const float*)d_in[26];
  const float* Wd1 = (const float*)d_in[27]; const float* bd1 = (const float*)d_in[28];
  const float* lng = (const float*)d_in[29]; const float* lnb = (const float*)d_in[30];
  const float* Wd2 = (const float*)d_in[31]; const float* bd2 = (const float*)d_in[32];
  const float* Wm  = (const float*)d_in[33]; const float* bm  = (const float*)d_in[34];
  const float* Wf1 = (const float*)d_in[35]; const float* bf1 = (const float*)d_in[36];
  const float* Wf2 = (const float*)d_in[37]; const float* bf2 = (const float*)d_in[38];
  const float* Wsp = (const float*)d_in[39]; const float* bsp = (const float*)d_in[40];
  const float* Wi  = (const float*)d_in[41]; const float* Wh  = (const float*)d_in[42];
  const float* bi  = (const float*)d_in[43]; const float* bh  = (const float*)d_in[44];
  const float* Wg  = (const float*)d_in[45]; const float* bg  = (const float*)d_in[46];

  float* out = (float*)d_out;
  float* o_predX   = out;
  float* o_predZ   = out + 23756800LL;
  float* o_tall    = out + 40140800LL;
  float* o_predS   = out + 40960000LL;
  float* o_predLG  = out + 41779200LL;
  float* o_predSig = out + 42598400LL;
  float* o_ctx     = out + 58982400LL;
  float* o_Y       = out + 59031552LL;
  float* o_times   = out + 68534272LL;
  float* o_mask    = out + 68861952LL;
  float* o_surv    = out + 78364672LL;
  float* o_dead    = out + 78692352LL;
  float* o_after   = out + 79020032LL;
  float* o_cens    = out + 79347712LL;
  float* o_wts     = out + 79364096LL;
  float* o_med     = out + 79380480LL;
  float* o_env     = out + 82657280LL;
  float* o_mu0     = out + 83083264LL;
  float* o_lv0     = out + 83410944LL;
  float* o_recon   = out + 83738624LL;
  float* o_mask0   = out + 84213760LL;

  float* ws    = (float*)d_ws;
  float* w_h1  = ws;
  float* w_h2  = ws + 1556480;
  float* w_z0  = ws + 2703360;
  float* w_hh  = ws + 3031040;
  float* w_cc  = ws + 3440640;
  float* w_sc1 = ws + 3850240;
  float* w_sh1 = w_sc1 + 95;
  float* w_sc2 = w_sh1 + 95;
  float* w_sh2 = w_sc2 + 70;

  // pass-through outputs (memory-bound; HBM-rate d2d copies)
  hipMemcpyAsync(o_Y,     Y,     sizeof(float)*9502720,  hipMemcpyDeviceToDevice, stream);
  hipMemcpyAsync(o_times, times, sizeof(float)*327680,   hipMemcpyDeviceToDevice, stream);
  hipMemcpyAsync(o_mask,  mask,  sizeof(float)*9502720,  hipMemcpyDeviceToDevice, stream);
  hipMemcpyAsync(o_surv,  surv,  sizeof(float)*327680,   hipMemcpyDeviceToDevice, stream);
  hipMemcpyAsync(o_dead,  dead,  sizeof(float)*327680,   hipMemcpyDeviceToDevice, stream);
  hipMemcpyAsync(o_after, after, sizeof(float)*327680,   hipMemcpyDeviceToDevice, stream);
  hipMemcpyAsync(o_cens,  cens,  sizeof(float)*16384,    hipMemcpyDeviceToDevice, stream);
  hipMemcpyAsync(o_wts,   wts,   sizeof(float)*16384,    hipMemcpyDeviceToDevice, stream);
  hipMemcpyAsync(o_med,   med,   sizeof(float)*3276800,  hipMemcpyDeviceToDevice, stream);
  hipMemcpyAsync(o_env,   env,   sizeof(float)*425984,   hipMemcpyDeviceToDevice, stream);
  hipMemcpyAsync(o_mask0, mask0, sizeof(float)*475136,   hipMemcpyDeviceToDevice, stream);

  enc1_kernel<<<BB/256, 256, 0, stream>>>(Y, times, mask0, env, med, missing, pop_std,
                                          eps_y0, We1, be1, w_h1);
  bnstats_kernel<<<95, 256, 0, stream>>>(w_h1, 95, bn1g, bn1b, w_sc1, w_sh1);
  enc2_kernel<<<BB/256, 256, 0, stream>>>(w_h1, w_sc1, w_sh1, We2, be2, w_h2);
  bnstats_kernel<<<70, 256, 0, stream>>>(w_h2, 70, bn2g, bn2b, w_sc2, w_sh2);
  enc3_kernel<<<BB/256, 256, 0, stream>>>(w_h2, w_sc2, w_sh2, We3, be3, eps_z, env, times,
                                          Wm, bm, o_mu0, o_lv0, o_ctx, w_z0, w_hh, w_cc);
  sde_kernel<<<BB/64, 128, 0, stream>>>(w_z0, w_hh, w_cc, times, o_ctx, eps_sde,
                                        Wf1, bf1, Wf2, bf2, Wsp, bsp, Wi, Wh, bi, bh,
                                        Wg, bg, o_predZ, o_tall, o_predS, o_predLG,
                                        o_predSig);
  dec_kernel<<<(BB*NT)/64, 128, 0, stream>>>(o_predZ, o_tall, o_ctx, Wd1, bd1, lng, lnb,
                                             Wd2, bd2, o_predX);
  recon_kernel<<<(BB*29 + 255)/256, 256, 0, stream>>>(o_predX, o_recon);
}